// SAGE_51908974739648
// MI455X (gfx1250) — compile-verified
//
#include <hip/hip_runtime.h>

#define N_NODES 100000
#define N_EDGES 640000
#define FDIM    128          // IN == HID == 128 (K is 128 for both layers)
#define OUTDIM  47
#define OUTPAD  48           // OUTDIM padded to a multiple of 16

typedef __attribute__((ext_vector_type(2))) float v2f;
typedef __attribute__((ext_vector_type(8))) float v8f;

// ---------------------------------------------------------------- utilities
__global__ void zero_f32(float* __restrict__ p, long n) {
  long i = (long)blockIdx.x * blockDim.x + threadIdx.x;
  long stride = (long)gridDim.x * blockDim.x;
  for (; i < n; i += stride) p[i] = 0.0f;
}

// deg[d] += 1 for every edge (float so the GEMM kernel can divide directly)
__global__ void degree_kernel(const int* __restrict__ dst,
                              float* __restrict__ deg) {
  int e = blockIdx.x * blockDim.x + threadIdx.x;
  if (e < N_EDGES) unsafeAtomicAdd(&deg[dst[e]], 1.0f);
}

// msg[dst[e]][f] += h[src[e]][f]   (one thread per (edge, feature))
// A wave's 32 threads share one edge -> src/dst broadcast from L0; the 512B
// feature row is read/updated fully coalesced. h (51MB) stays L2-resident.
__global__ void scatter_kernel(const float* __restrict__ h,
                               const int* __restrict__ src,
                               const int* __restrict__ dst,
                               float* __restrict__ msg) {
  long gid = (long)blockIdx.x * blockDim.x + threadIdx.x;
  if (gid >= (long)N_EDGES * FDIM) return;
  int e = (int)(gid >> 7);
  int f = (int)(gid & (FDIM - 1));
  int s = src[e];
  int d = dst[e];
  unsafeAtomicAdd(&msg[d * FDIM + f], h[s * FDIM + f]);
}

// Wt[n][k] = (n < N) ? W[k][n] : 0   -- transposed + column-padded weights so
// the GEMM reads each lane's B column contiguously along k (b64 loads,
// immediate offsets, no divergent guards in the inner loop).
__global__ void transpose_pad_kernel(const float* __restrict__ W,
                                     float* __restrict__ Wt,
                                     int N, int Npad) {
  int idx = blockIdx.x * blockDim.x + threadIdx.x;
  if (idx >= Npad * FDIM) return;
  int n = idx >> 7;          // / FDIM
  int k = idx & (FDIM - 1);  // % FDIM
  Wt[n * FDIM + k] = (n < N) ? W[k * N + n] : 0.0f;
}

// ------------------------------------------------------------- fused GEMM
// out[16 x NOUT] tile = ReLU?( Aself@Wself + (Amsg/deg)@Wneigh + bias )
// One block per 16-row M tile, one wave per 16-col N tile (NPAD/16 waves).
// V_WMMA_F32_16X16X4_F32: full fp32 precision, matches the reference.
#define LDS_STRIDE 132   // 16 rows padded: lane row-stride hits distinct banks

template <int NOUT, int NPAD, bool RELU>
__global__ void sage_gemm_wmma(const float* __restrict__ Aself,
                               const float* __restrict__ Amsg,
                               const float* __restrict__ deg,
                               const float* __restrict__ WtS,   // [NPAD][128]
                               const float* __restrict__ WtN,   // [NPAD][128]
                               const float* __restrict__ bias,
                               float* __restrict__ out) {
  __shared__ float ldsS[16 * LDS_STRIDE];
  __shared__ float ldsN[16 * LDS_STRIDE];

  const int mbase = blockIdx.x * 16;   // 6250 * 16 == 100000 exactly

  // Cooperative staging of both 16x128 A tiles (coalesced b32 loads),
  // applying the mean normalization (msg / max(deg,1)) on the way in.
  for (int i = threadIdx.x; i < 16 * FDIM; i += blockDim.x) {
    int r = i >> 7;
    int c = i & (FDIM - 1);
    int row = mbase + r;
    float dv = fmaxf(deg[row], 1.0f);
    ldsS[r * LDS_STRIDE + c] = Aself[row * FDIM + c];
    ldsN[r * LDS_STRIDE + c] = Amsg[row * FDIM + c] / dv;
  }
  __syncthreads();

  const int lane  = threadIdx.x & 31;
  const int wave  = threadIdx.x >> 5;
  const int nbase = wave * 16;

  // f32 16x16x4 VGPR layout (ISA 7.12.2):
  //  A: lane m=lane&15 holds (k+ksub, k+ksub+1), ksub = (lane>>4)*2
  //  B: same K split across lane halves, n = lane&15
  const int mrow = lane & 15;
  const int ksub = (lane >> 4) << 1;
  const int ncol = nbase + (lane & 15);          // always < NPAD

  const float* __restrict__ As = ldsS + mrow * LDS_STRIDE + ksub;
  const float* __restrict__ An = ldsN + mrow * LDS_STRIDE + ksub;
  const float* __restrict__ Bs = WtS + ncol * FDIM + ksub;   // contiguous in k
  const float* __restrict__ Bn = WtN + ncol * FDIM + ksub;

  v8f acc = {};
  #pragma unroll 8
  for (int k = 0; k < FDIM; k += 4) {
    v2f a = *(const v2f*)(As + k);     // ds_load_b64
    v2f b = *(const v2f*)(Bs + k);     // global_load_b64, immediate offset
    acc = __builtin_amdgcn_wmma_f32_16x16x4_f32(
        false, a, false, b, (short)0, acc, false, false);
  }
  #pragma unroll 8
  for (int k = 0; k < FDIM; k += 4) {
    v2f a = *(const v2f*)(An + k);
    v2f b = *(const v2f*)(Bn + k);
    acc = __builtin_amdgcn_wmma_f32_16x16x4_f32(
        false, a, false, b, (short)0, acc, false, false);
  }

  // C/D layout: VGPR r -> M = mbase + r + (lane>>4)*8, N = ncol
  if (NOUT == NPAD || ncol < NOUT) {
    float bv = bias[ncol];
    const int moff = (lane >> 4) << 3;
    #pragma unroll
    for (int r = 0; r < 8; ++r) {
      int m = mbase + r + moff;
      float v = acc[r] + bv;
      if (RELU) v = fmaxf(v, 0.0f);
      out[m * NOUT + ncol] = v;
    }
  }
}

// ---------------------------------------------------------------- launcher
extern "C" void kernel_launch(void* const* d_in, const int* in_sizes, int n_in,
                              void* d_out, int out_size, void* d_ws, size_t ws_size,
                              hipStream_t stream) {
  const float* features = (const float*)d_in[0];
  const int*   src      = (const int*)  d_in[1];
  const int*   dst      = (const int*)  d_in[2];
  const float* w_self1  = (const float*)d_in[3];
  const float* w_neigh1 = (const float*)d_in[4];
  const float* b1       = (const float*)d_in[5];
  const float* w_self2  = (const float*)d_in[6];
  const float* w_neigh2 = (const float*)d_in[7];
  const float* b2       = (const float*)d_in[8];
  float* out = (float*)d_out;

  // workspace: deg[N] | msg[N*128] (reused) | h1[N*128] | 4 transposed weights
  float* deg  = (float*)d_ws;
  float* msg  = deg + N_NODES;
  float* h1   = msg + (size_t)N_NODES * FDIM;
  float* wtS1 = h1  + (size_t)N_NODES * FDIM;   // [128][128]
  float* wtN1 = wtS1 + FDIM * FDIM;             // [128][128]
  float* wtS2 = wtN1 + FDIM * FDIM;             // [48][128] zero-padded
  float* wtN2 = wtS2 + OUTPAD * FDIM;           // [48][128] zero-padded

  const long edge_threads   = (long)N_EDGES * FDIM;
  const int  scatter_blocks = (int)((edge_threads + 255) / 256);

  // weight prep (tiny; L2-resident thereafter)
  transpose_pad_kernel<<<(FDIM * FDIM + 255) / 256, 256, 0, stream>>>(
      w_self1, wtS1, FDIM, FDIM);
  transpose_pad_kernel<<<(FDIM * FDIM + 255) / 256, 256, 0, stream>>>(
      w_neigh1, wtN1, FDIM, FDIM);
  transpose_pad_kernel<<<(OUTPAD * FDIM + 255) / 256, 256, 0, stream>>>(
      w_self2, wtS2, OUTDIM, OUTPAD);
  transpose_pad_kernel<<<(OUTPAD * FDIM + 255) / 256, 256, 0, stream>>>(
      w_neigh2, wtN2, OUTDIM, OUTPAD);

  // zero deg + msg (contiguous N*(1+128) floats)
  zero_f32<<<4096, 256, 0, stream>>>(deg, (long)N_NODES * (1 + FDIM));
  degree_kernel<<<(N_EDGES + 255) / 256, 256, 0, stream>>>(dst, deg);

  // ---- layer 1 ----
  scatter_kernel<<<scatter_blocks, 256, 0, stream>>>(features, src, dst, msg);
  sage_gemm_wmma<FDIM, FDIM, true><<<N_NODES / 16, 256, 0, stream>>>(
      features, msg, deg, wtS1, wtN1, b1, h1);

  // ---- layer 2 ----
  zero_f32<<<4096, 256, 0, stream>>>(msg, (long)N_NODES * FDIM);
  scatter_kernel<<<scatter_blocks, 256, 0, stream>>>(h1, src, dst, msg);
  // OUTPAD=48 -> 3 N-tiles -> 3 waves (96 threads) per block
  sage_gemm_wmma<OUTDIM, OUTPAD, false><<<N_NODES / 16, 96, 0, stream>>>(
      h1, msg, deg, wtS2, wtN2, b2, out);
}